// PyramidScaledDotProductAttention_73942156968094
// MI455X (gfx1250) — compile-verified
//
#include <hip/hip_runtime.h>

// Scaled dot-product attention for MI455X (gfx1250, wave32).
//   B=256, T=512, D=64, temperature = 8.0
// Outputs (concatenated in d_out): output [B,T,D] f32, attn [B,T,T] f32.
//
// Roofline: attn write = 268 MB dominates (23.3 TB/s -> ~17us floor), so keep
// full fp32 precision and use V_WMMA_F32_16X16X4_F32 for both GEMMs.
// One wave32 per 16-row query tile; scores staged in LDS (conflict-free
// padded strides); attn written with b128 stores.

typedef float v2f __attribute__((ext_vector_type(2)));
typedef float v8f __attribute__((ext_vector_type(8)));

#define B_ 256
#define T_ 512
#define D_ 64
#define SCALE 0.125f      // 1 / temperature

#define QK_STRIDE 68      // padded LDS row stride (floats) for 64-col tiles
#define S_STRIDE  516     // padded LDS row stride (floats) for 512-col score rows

__launch_bounds__(32, 1)
__global__ void PyramidScaledDotProductAttention_73942156968094_kernel(
    const float* __restrict__ Q, const float* __restrict__ K,
    const float* __restrict__ V, float* __restrict__ Out,
    float* __restrict__ Attn) {
  // sS doubles as the Q staging buffer (Q tile is dead once qa[] is in regs).
  __shared__ float sS[16 * S_STRIDE];    // scores/probs 16x512 (+Q staging)
  __shared__ float sKV[16 * QK_STRIDE];  // K/V tile 16x64 (reused)

  const int lane = threadIdx.x;            // 0..31, single wave per block
  const int b    = blockIdx.x >> 5;        // batch
  const int q0   = (blockIdx.x & 31) << 4; // query-tile base row

  const int m16 = lane & 15;               // row/col index for A/B layouts
  const int hi  = lane >> 4;               // half-wave select
  const int kb2 = hi * 2;                  // K-subcolumn base (A/B layouts)

  const size_t bq = (size_t)b * T_ + q0;
  const float* Qb = Q + bq * D_;
  const float* Kb = K + (size_t)b * T_ * D_;
  const float* Vb = V + (size_t)b * T_ * D_;

  // ---- stage Q tile (16x64) into LDS front of sS, coalesced 16B loads ----
  for (int i = lane; i < 16 * 16; i += 32) {
    const int r = i >> 4, c = (i & 15) << 2;
    *(float4*)&sS[r * QK_STRIDE + c] = *(const float4*)(Qb + (size_t)r * D_ + c);
  }
  __syncthreads();  // single-wave WG: lowers to wave-local ordering (free)

  // ---- preload Q in WMMA A-layout: qa[j] = Q[m16][4j+kb2 .. +1] ----
  v2f qa[16];
#pragma unroll
  for (int j = 0; j < 16; ++j)
    qa[j] = *(const v2f*)&sS[m16 * QK_STRIDE + 4 * j + kb2];
  __syncthreads();

  // ================= S = Q @ K^T * (1/temp) =================
  for (int n0 = 0; n0 < T_; n0 += 16) {
    for (int i = lane; i < 16 * 16; i += 32) {
      const int r = i >> 4, c = (i & 15) << 2;
      *(float4*)&sKV[r * QK_STRIDE + c] =
          *(const float4*)(Kb + (size_t)(n0 + r) * D_ + c);
    }
    if (n0 + 16 < T_)  // prefetch next K tile -> global_prefetch_b8
      __builtin_prefetch(Kb + (size_t)(n0 + 16 + m16) * D_ + hi * 32, 0, 1);
    __syncthreads();

    // two independent D->C accumulation chains to overlap WMMA latency
    v8f acc0 = {}, acc1 = {};
#pragma unroll
    for (int j = 0; j < 16; j += 2) {
      // B-layout of K^T tile: lane n holds K[n0+n][4j+kb2 .. +1]
      const v2f kv0 = *(const v2f*)&sKV[m16 * QK_STRIDE + 4 * j + kb2];
      const v2f kv1 = *(const v2f*)&sKV[m16 * QK_STRIDE + 4 * (j + 1) + kb2];
      acc0 = __builtin_amdgcn_wmma_f32_16x16x4_f32(
          false, qa[j], false, kv0, (short)0, acc0, false, false);
      acc1 = __builtin_amdgcn_wmma_f32_16x16x4_f32(
          false, qa[j + 1], false, kv1, (short)0, acc1, false, false);
    }
    const v8f acc = acc0 + acc1;
    // C layout: VGPR r -> row (r + 8*hi), col = lane&15
#pragma unroll
    for (int r = 0; r < 8; ++r)
      sS[(r + 8 * hi) * S_STRIDE + n0 + m16] = acc[r] * SCALE;
    __syncthreads();
  }
  __syncthreads();

  // ================= row softmax + attn write (b128) =================
  float* attn_base = Attn + bq * T_;
  for (int m = 0; m < 16; ++m) {
    float4 vals[4];
    float mx = -3.402823466e38f;
#pragma unroll
    for (int i = 0; i < 4; ++i) {
      vals[i] = *(const float4*)&sS[m * S_STRIDE + 4 * lane + 128 * i];
      mx = fmaxf(mx, fmaxf(fmaxf(vals[i].x, vals[i].y),
                           fmaxf(vals[i].z, vals[i].w)));
    }
#pragma unroll
    for (int off = 16; off > 0; off >>= 1)
      mx = fmaxf(mx, __shfl_xor(mx, off, 32));
    float sum = 0.f;
#pragma unroll
    for (int i = 0; i < 4; ++i) {
      vals[i].x = __expf(vals[i].x - mx);
      vals[i].y = __expf(vals[i].y - mx);
      vals[i].z = __expf(vals[i].z - mx);
      vals[i].w = __expf(vals[i].w - mx);
      sum += (vals[i].x + vals[i].y) + (vals[i].z + vals[i].w);
    }
#pragma unroll
    for (int off = 16; off > 0; off >>= 1)
      sum += __shfl_xor(sum, off, 32);
    const float inv = 1.0f / sum;
    float* arow = attn_base + (size_t)m * T_;
#pragma unroll
    for (int i = 0; i < 4; ++i) {
      float4 p = vals[i];
      p.x *= inv; p.y *= inv; p.z *= inv; p.w *= inv;
      *(float4*)&sS[m * S_STRIDE + 4 * lane + 128 * i] = p;  // keep P for P@V
      *(float4*)&arow[4 * lane + 128 * i] = p;               // global_store_b128
    }
  }
  __syncthreads();

  // ================= O = P @ V =================
  v8f o0 = {}, o1 = {}, o2 = {}, o3 = {};
  for (int k0 = 0; k0 < T_; k0 += 16) {
    for (int i = lane; i < 16 * 16; i += 32) {
      const int r = i >> 4, c = (i & 15) << 2;
      *(float4*)&sKV[r * QK_STRIDE + c] =
          *(const float4*)(Vb + (size_t)(k0 + r) * D_ + c);
    }
    if (k0 + 16 < T_)  // prefetch next V tile
      __builtin_prefetch(Vb + (size_t)(k0 + 16 + m16) * D_ + hi * 32, 0, 1);
    __syncthreads();
#pragma unroll
    for (int jj = 0; jj < 4; ++jj) {
      const int kk = jj * 4 + kb2;  // per-lane K row within this V tile
      // A-layout of P: lane m16 holds P[m16][k0+kk .. +1]
      const v2f pa = *(const v2f*)&sS[m16 * S_STRIDE + k0 + kk];
      const float* vr0 = &sKV[(kk + 0) * QK_STRIDE];
      const float* vr1 = &sKV[(kk + 1) * QK_STRIDE];
      {
        const v2f vb = {vr0[0 + m16], vr1[0 + m16]};
        o0 = __builtin_amdgcn_wmma_f32_16x16x4_f32(false, pa, false, vb,
                                                   (short)0, o0, false, false);
      }
      {
        const v2f vb = {vr0[16 + m16], vr1[16 + m16]};
        o1 = __builtin_amdgcn_wmma_f32_16x16x4_f32(false, pa, false, vb,
                                                   (short)0, o1, false, false);
      }
      {
        const v2f vb = {vr0[32 + m16], vr1[32 + m16]};
        o2 = __builtin_amdgcn_wmma_f32_16x16x4_f32(false, pa, false, vb,
                                                   (short)0, o2, false, false);
      }
      {
        const v2f vb = {vr0[48 + m16], vr1[48 + m16]};
        o3 = __builtin_amdgcn_wmma_f32_16x16x4_f32(false, pa, false, vb,
                                                   (short)0, o3, false, false);
      }
    }
    __syncthreads();
  }

  // store output tile: C layout -> row (r + 8*hi), col (16t + m16)
  float* orow = Out + bq * D_;
#pragma unroll
  for (int r = 0; r < 8; ++r) {
    const size_t m = (size_t)(r + 8 * hi);
    orow[m * D_ +  0 + m16] = o0[r];
    orow[m * D_ + 16 + m16] = o1[r];
    orow[m * D_ + 32 + m16] = o2[r];
    orow[m * D_ + 48 + m16] = o3[r];
  }
}

extern "C" void kernel_launch(void* const* d_in, const int* in_sizes, int n_in,
                              void* d_out, int out_size, void* d_ws, size_t ws_size,
                              hipStream_t stream) {
  const float* q = (const float*)d_in[0];
  const float* k = (const float*)d_in[1];
  const float* v = (const float*)d_in[2];
  float* out  = (float*)d_out;                       // [B,T,D]
  float* attn = out + (size_t)B_ * T_ * D_;          // [B,T,T]
  const int grid = B_ * (T_ / 16);                   // 8192 blocks, 1 wave each
  PyramidScaledDotProductAttention_73942156968094_kernel<<<grid, 32, 0, stream>>>(
      q, k, v, out, attn);
}